// RiemannCompressedLinear_43069932044591
// MI455X (gfx1250) — compile-verified
//
#include <hip/hip_runtime.h>

typedef float v2f __attribute__((ext_vector_type(2)));
typedef float v8f __attribute__((ext_vector_type(8)));

#define IN_F   4096
#define OUT_F  4096
#define RANK   64
#define NROWS  8192                   // B*S = 4*2048
#define KCHUNKS 4
#define KCHUNK_LEN (IN_F / KCHUNKS)   // 1024
#define KGROUPS (IN_F / 4)            // 1024 k-groups of 4

static __device__ __forceinline__ v8f splat8(float v) {
    v8f r;
#pragma unroll
    for (int j = 0; j < 8; ++j) r[j] = v;
    return r;
}

// -------------------------------------------------------------------------
// Prepack V [4096,64] into B-fragment order:
//   Vp[g][t][lane][0..1] = { V[4g + kb][16t + m], V[4g + kb + 1][16t + m] }
// with m = lane&15, kb = (lane>>4)*2. One b64 per lane per (g,t) in stage 1,
// consecutive lanes -> consecutive 8B -> fully coalesced.
// -------------------------------------------------------------------------
__global__ __launch_bounds__(256) void lr_prepack_v(const float* __restrict__ V,
                                                    float* __restrict__ Vp) {
    const int idx  = blockIdx.x * 256 + threadIdx.x;  // 0 .. 131071
    const int lane = idx & 31;
    const int t    = (idx >> 5) & 3;
    const int g    = idx >> 7;                        // 0..1023
    const int m    = lane & 15;
    const int kb   = (lane >> 4) << 1;
    const int krow = g * 4 + kb;
    v2f b = { V[(size_t)krow * RANK + t * 16 + m],
              V[(size_t)(krow + 1) * RANK + t * 16 + m] };
    *(v2f*)(Vp + (size_t)idx * 2) = b;
}

// -------------------------------------------------------------------------
// Stage 1:  Tpart[p] = x[:, p*1024:(p+1)*1024] @ V[p*1024:(p+1)*1024, :]
// One wave per (16-row tile, k-chunk). 512 row tiles * 4 chunks = 2048 waves.
// A fragment (16x4 f32): lane m = lane&15 is the row, half-wave picks K pair.
// -------------------------------------------------------------------------
__global__ __launch_bounds__(128) void lr_stage1(const float* __restrict__ x,
                                                 const float* __restrict__ Vp,
                                                 float* __restrict__ Tpart) {
    const int wave     = blockIdx.x * 4 + (threadIdx.x >> 5); // 4 waves/block
    const int row_tile = wave >> 2;                           // 0..511
    const int kchunk   = wave & 3;                            // 0..3
    const int lane     = threadIdx.x & 31;
    const int m        = lane & 15;
    const int hi       = lane >> 4;
    const int kb       = hi << 1;                             // 0 or 2

    const float* xrow = x + (size_t)(row_tile * 16 + m) * IN_F + kb;

    v8f acc0 = {}, acc1 = {}, acc2 = {}, acc3 = {};

    const int g0 = kchunk * (KCHUNK_LEN / 4);                 // 256 groups/chunk
#pragma unroll 4
    for (int g = g0; g < g0 + KCHUNK_LEN / 4; ++g) {
        // A: x[row, 4g+kb : 4g+kb+2]  (8-byte aligned global_load_b64)
        v2f a = *(const v2f*)(xrow + g * 4);
        // B: prepacked fragments, one b64 per tile, imm offsets 0/256/512/768B
        const float* vg = Vp + (size_t)g * 128 + lane * 2;
        v2f b0 = *(const v2f*)(vg);
        v2f b1 = *(const v2f*)(vg + 64);
        v2f b2 = *(const v2f*)(vg + 128);
        v2f b3 = *(const v2f*)(vg + 192);
        acc0 = __builtin_amdgcn_wmma_f32_16x16x4_f32(false, a, false, b0, (short)0, acc0, false, false);
        acc1 = __builtin_amdgcn_wmma_f32_16x16x4_f32(false, a, false, b1, (short)0, acc1, false, false);
        acc2 = __builtin_amdgcn_wmma_f32_16x16x4_f32(false, a, false, b2, (short)0, acc2, false, false);
        acc3 = __builtin_amdgcn_wmma_f32_16x16x4_f32(false, a, false, b3, (short)0, acc3, false, false);
    }

    // D layout: VGPR j holds row (j + 8*hi), column = m
    float* Tp = Tpart + (size_t)kchunk * NROWS * RANK;
#pragma unroll
    for (int j = 0; j < 8; ++j) {
        float* dst = Tp + (size_t)(row_tile * 16 + hi * 8 + j) * RANK + m;
        dst[0]  = acc0[j];
        dst[16] = acc1[j];
        dst[32] = acc2[j];
        dst[48] = acc3[j];
    }
}

// -------------------------------------------------------------------------
// Stage 2:  out = (sum_p Tpart[p]) @ U^T + bias
// One wave per 16-row x 64-col output strip: 512*64 = 32768 waves.
// B[k][n] = U[n][k]  -> contiguous float2 loads from row-major U.
// Accumulators seeded with bias (broadcast over rows).
// -------------------------------------------------------------------------
__global__ __launch_bounds__(256) void lr_stage2(const float* __restrict__ Tpart,
                                                 const float* __restrict__ U,
                                                 const float* __restrict__ bias,
                                                 float* __restrict__ out) {
    const int wave     = blockIdx.x * 8 + (threadIdx.x >> 5); // 8 waves/block
    const int row_tile = wave >> 6;                           // 0..511
    const int colg     = wave & 63;                           // 0..63
    const int lane     = threadIdx.x & 31;
    const int m        = lane & 15;
    const int hi       = lane >> 4;
    const int kb       = hi << 1;
    const int n0       = colg * 64;

    v8f acc0 = splat8(bias[n0 + m]);
    v8f acc1 = splat8(bias[n0 + 16 + m]);
    v8f acc2 = splat8(bias[n0 + 32 + m]);
    v8f acc3 = splat8(bias[n0 + 48 + m]);

    const float* trow = Tpart + (size_t)(row_tile * 16 + m) * RANK + kb;
    const size_t pstride = (size_t)NROWS * RANK;
    const float* u0 = U + (size_t)(n0 + m) * RANK + kb;

#pragma unroll
    for (int k = 0; k < RANK; k += 4) {
        // A: sum of the 4 K-chunk partials (linearity of matmul)
        v2f a  = *(const v2f*)(trow + k);
        v2f ap = *(const v2f*)(trow + pstride     + k);
        a = a + ap;
        ap = *(const v2f*)(trow + 2 * pstride + k);
        a = a + ap;
        ap = *(const v2f*)(trow + 3 * pstride + k);
        a = a + ap;
        // B: U[n][k+kb : k+kb+2] for the 4 column tiles
        v2f b0 = *(const v2f*)(u0 +                    k);
        v2f b1 = *(const v2f*)(u0 + (size_t)16 * RANK + k);
        v2f b2 = *(const v2f*)(u0 + (size_t)32 * RANK + k);
        v2f b3 = *(const v2f*)(u0 + (size_t)48 * RANK + k);
        acc0 = __builtin_amdgcn_wmma_f32_16x16x4_f32(false, a, false, b0, (short)0, acc0, false, false);
        acc1 = __builtin_amdgcn_wmma_f32_16x16x4_f32(false, a, false, b1, (short)0, acc1, false, false);
        acc2 = __builtin_amdgcn_wmma_f32_16x16x4_f32(false, a, false, b2, (short)0, acc2, false, false);
        acc3 = __builtin_amdgcn_wmma_f32_16x16x4_f32(false, a, false, b3, (short)0, acc3, false, false);
    }

    // Streaming (non-temporal) stores: out is never re-read, keep L2 for U/T/Vp
#pragma unroll
    for (int j = 0; j < 8; ++j) {
        float* dst = out + (size_t)(row_tile * 16 + hi * 8 + j) * OUT_F + n0 + m;
        __builtin_nontemporal_store(acc0[j], dst);
        __builtin_nontemporal_store(acc1[j], dst + 16);
        __builtin_nontemporal_store(acc2[j], dst + 32);
        __builtin_nontemporal_store(acc3[j], dst + 48);
    }
}

extern "C" void kernel_launch(void* const* d_in, const int* in_sizes, int n_in,
                              void* d_out, int out_size, void* d_ws, size_t ws_size,
                              hipStream_t stream) {
    const float* x    = (const float*)d_in[0];   // [8192, 4096]
    const float* U    = (const float*)d_in[1];   // [4096, 64]
    const float* V    = (const float*)d_in[2];   // [4096, 64]
    const float* bias = (const float*)d_in[3];   // [4096]
    float* out   = (float*)d_out;                // [8192, 4096]

    float* Tpart = (float*)d_ws;                                   // 8 MB
    float* Vp    = (float*)d_ws + (size_t)KCHUNKS * NROWS * RANK;  // +1 MB

    // Prepack V into B-fragment order (1 MB, trivial)
    lr_prepack_v<<<512, 256, 0, stream>>>(V, Vp);
    // Stage 1: 512 row tiles * 4 k-chunks = 2048 waves, 4 waves/block
    lr_stage1<<<512, 128, 0, stream>>>(x, Vp, Tpart);
    // Stage 2: 512 row tiles * 64 col groups = 32768 waves, 8 waves/block
    lr_stage2<<<4096, 256, 0, stream>>>(Tpart, U, bias, out);
}